// GCN_62139586839308
// MI455X (gfx1250) — compile-verified
//
#include <hip/hip_runtime.h>
#include <stdint.h>

// GCN propagation on bipartite COO graph (LightGCN-style, no weight GEMMs).
// Output layout (row-major): users [U, 192] then items [I, 192];
// cols [0:64)=emb, [64:128)=gcn1, [128:192)=gcn2.
//
// Roofline: 4 SpMMs x 2*E*F ~ 0.82 GFLOP vs ~3.4 GB gather/atomic traffic
// -> pure memory/atomic problem. Working set (~172 MB) fits the 192 MB L2,
// so f32 atomics resolve at the L2 atomic units; HBM sees only compulsory
// traffic. WMMA is inapplicable (arbitrary-row scatter/gather); the CDNA5
// feature that fits is the async global->LDS path for edge metadata.

#define EPB 16          // edges per 256-thread block (16 lanes per edge, f32x4 each)
#define OUT_STRIDE 192  // 3*F

typedef __attribute__((ext_vector_type(4))) float f32x4;  // native vector (NT-store capable)

__device__ __forceinline__ f32x4 ldg4(const float* p) {
    return *reinterpret_cast<const f32x4*>(p);
}

// ---------------------------------------------------------------------------
// Kernel 1: seed output. cols[0:64)=emb (pure copy, never re-read -> NT
// stores so it doesn't pollute L2), cols[64:128)=emb (gcn1 accumulator
// starts at "+ embedding", RT so it stays L2-resident for the atomics).
// ---------------------------------------------------------------------------
__global__ __launch_bounds__(256) void gcn_init(
    const float* __restrict__ uemb, const float* __restrict__ iemb,
    float* __restrict__ outU, float* __restrict__ outI, int U, int I)
{
    const size_t tid = (size_t)blockIdx.x * 256 + threadIdx.x;  // one f32x4 chunk
    const size_t nU = (size_t)U * 16;
    const size_t nI = (size_t)I * 16;
    if (tid < nU) {
        const size_t r = tid >> 4; const int c = (int)(tid & 15);
        const f32x4 v = ldg4(uemb + r * 64 + c * 4);
        __builtin_nontemporal_store(v, reinterpret_cast<f32x4*>(outU + r * OUT_STRIDE + c * 4));
        *reinterpret_cast<f32x4*>(outU + r * OUT_STRIDE + 64 + c * 4) = v;
    } else if (tid < nU + nI) {
        const size_t q = tid - nU;
        const size_t r = q >> 4; const int c = (int)(q & 15);
        const f32x4 v = ldg4(iemb + r * 64 + c * 4);
        __builtin_nontemporal_store(v, reinterpret_cast<f32x4*>(outI + r * OUT_STRIDE + c * 4));
        *reinterpret_cast<f32x4*>(outI + r * OUT_STRIDE + 64 + c * 4) = v;
    }
}

// ---------------------------------------------------------------------------
// Kernel 3: carry gcn1 -> gcn2 accumulator (cols 64:128 -> 128:192).
// outU/outI are contiguous with identical stride, so treat as U+I rows.
// ---------------------------------------------------------------------------
__global__ __launch_bounds__(256) void gcn_carry(
    float* __restrict__ out, size_t nRows)
{
    const size_t tid = (size_t)blockIdx.x * 256 + threadIdx.x;
    if (tid < nRows * 16) {
        const size_t r = tid >> 4; const int c = (int)(tid & 15);
        const f32x4 v = *reinterpret_cast<const f32x4*>(out + r * OUT_STRIDE + 64 + c * 4);
        *reinterpret_cast<f32x4*>(out + r * OUT_STRIDE + 128 + c * 4) = v;
    }
}

// ---------------------------------------------------------------------------
// Kernels 2 & 4: edge scatter pass. Each 256-thread block handles 16 edges;
// each wave32 owns 2 edges (16 lanes x f32x4 slice per edge).
//   accU[edge_user] += ui_val * gatherI[edge_item]   (spmm_ui term)
//   accI[edge_item] += iu_val * gatherU[edge_user]   (spmm_iu term)
// Each wave stages its own 2 edges' metadata (8 dwords) into a private LDS
// slice via CDNA5 async global->LDS copies (ASYNCcnt), then consumes after
// s_wait_asynccnt 0 -- no workgroup barrier needed (per-wave counter,
// completion == data written to LDS).
// ---------------------------------------------------------------------------
__global__ __launch_bounds__(256) void gcn_edge_pass(
    const float* __restrict__ gatherU, int strideU,
    const float* __restrict__ gatherI, int strideI,
    const float* __restrict__ ui_vals,
    const float* __restrict__ iu_vals,
    const int* __restrict__ edge_user,
    const int* __restrict__ edge_item,
    float* __restrict__ accU,   // pre-offset to destination column, stride 192
    float* __restrict__ accI,   // pre-offset to destination column, stride 192
    int E)
{
    __shared__ uint32_t sh[8 * 8];       // 8 waves x [u0,u1,i0,i1,uv0,uv1,iv0,iv1]
    const int t    = threadIdx.x;
    const int wave = t >> 5;
    const int lane = t & 31;
    const int ew   = blockIdx.x * EPB + wave * 2;   // first edge owned by this wave

    uint32_t* wsh = sh + wave * 8;
    if (lane < 8) {
        const int a = lane >> 1;         // 0=user, 1=item, 2=ui_val, 3=iu_val
        const int k = lane & 1;          // edge 0/1 within wave
        const int e = ew + k;
        const uint32_t* gp =
            (a == 0) ? reinterpret_cast<const uint32_t*>(edge_user) + e :
            (a == 1) ? reinterpret_cast<const uint32_t*>(edge_item) + e :
            (a == 2) ? reinterpret_cast<const uint32_t*>(ui_vals)   + e :
                       reinterpret_cast<const uint32_t*>(iu_vals)   + e;
        if (e < E) {
            // Low 32 bits of the flat shared address == LDS byte address.
            const uint32_t lds = (uint32_t)(uintptr_t)(&wsh[lane]);
            asm volatile("global_load_async_to_lds_b32 %0, %1, off"
                         :: "v"(lds), "v"(gp) : "memory");
        }
    }
    asm volatile("s_wait_asynccnt 0x0" ::: "memory");

    const int g  = lane >> 4;            // edge slot within wave (0/1)
    const int lg = lane & 15;            // f32x4 feature slice
    const int e  = ew + g;
    if (e >= E) return;

    const int   u  = (int)wsh[0 + g];
    const int   it = (int)wsh[2 + g];
    const float uv = __uint_as_float(wsh[4 + g]);
    const float iv = __uint_as_float(wsh[6 + g]);

    const f32x4 xu = ldg4(gatherU + (size_t)u  * strideU + lg * 4);
    const f32x4 xi = ldg4(gatherI + (size_t)it * strideI + lg * 4);

    float* du = accU + (size_t)u  * OUT_STRIDE + lg * 4;
    float* di = accI + (size_t)it * OUT_STRIDE + lg * 4;
    atomicAdd(du + 0, uv * xi.x);
    atomicAdd(du + 1, uv * xi.y);
    atomicAdd(du + 2, uv * xi.z);
    atomicAdd(du + 3, uv * xi.w);
    atomicAdd(di + 0, iv * xu.x);
    atomicAdd(di + 1, iv * xu.y);
    atomicAdd(di + 2, iv * xu.z);
    atomicAdd(di + 3, iv * xu.w);
}

// ---------------------------------------------------------------------------
extern "C" void kernel_launch(void* const* d_in, const int* in_sizes, int n_in,
                              void* d_out, int out_size, void* d_ws, size_t ws_size,
                              hipStream_t stream)
{
    const float* uemb = (const float*)d_in[0];
    const float* iemb = (const float*)d_in[1];
    const float* uiv  = (const float*)d_in[2];
    const float* iuv  = (const float*)d_in[3];
    const int*   eu   = (const int*)d_in[4];
    const int*   ei   = (const int*)d_in[5];

    const int U = in_sizes[0] / 64;
    const int I = in_sizes[1] / 64;
    const int E = in_sizes[2];

    float* outU = (float*)d_out;
    float* outI = outU + (size_t)U * OUT_STRIDE;

    // 1) seed emb into concat slot 0 (NT) and gcn1 accumulator (RT)
    {
        const size_t n = (size_t)(U + I) * 16;
        gcn_init<<<(int)((n + 255) / 256), 256, 0, stream>>>(uemb, iemb, outU, outI, U, I);
    }
    // 2) layer-1 edges: gather original embeddings (stride 64), scatter into col 64
    {
        const int blocks = (E + EPB - 1) / EPB;
        gcn_edge_pass<<<blocks, 256, 0, stream>>>(uemb, 64, iemb, 64,
                                                  uiv, iuv, eu, ei,
                                                  outU + 64, outI + 64, E);
    }
    // 3) gcn2 accumulator starts at gcn1
    {
        const size_t n = (size_t)(U + I) * 16;
        gcn_carry<<<(int)((n + 255) / 256), 256, 0, stream>>>(outU, (size_t)(U + I));
    }
    // 4) layer-2 edges: gather gcn1 (stride 192, col 64), scatter into col 128
    {
        const int blocks = (E + EPB - 1) / EPB;
        gcn_edge_pass<<<blocks, 256, 0, stream>>>(outU + 64, OUT_STRIDE,
                                                  outI + 64, OUT_STRIDE,
                                                  uiv, iuv, eu, ei,
                                                  outU + 128, outI + 128, E);
    }
    (void)d_ws; (void)ws_size; (void)n_in; (void)out_size;
}